// TorchBF_59107339928064
// MI455X (gfx1250) — compile-verified
//
#include <hip/hip_runtime.h>
#include <hip/hip_bf16.h>

typedef float v2f __attribute__((ext_vector_type(2)));
typedef float v8f __attribute__((ext_vector_type(8)));

#define KSPK 2
#define MMSK 2
#define T_   3000
#define F_   513
#define D_   6
#define TF_  (T_ * F_)
#define NCHUNK 8
#define TCHUNK ((T_ + NCHUNK - 1) / NCHUNK)   // 375
#define TT   16                                // time tile per LDS buffer
#define TPAD 17                                // padded t-stride: 17*16 = 272 = 16 mod 64 -> conflict-free
#define PARTIAL_FLOATS ((size_t)NCHUNK * MMSK * KSPK * F_ * 36 * 2)

// One async Global->LDS dword copy (GVS mode: SGPR base + 32-bit byte offset).
// Tracked by ASYNCcnt; per-wave issue count is uniform (callers never predicate).
__device__ __forceinline__ void async_ld(unsigned lds_off, const void* base, int byte_off) {
  asm volatile("global_load_async_to_lds_b32 %0, %1, %2"
               :: "v"(lds_off), "v"(byte_off), "s"(base) : "memory");
}
#define ASYNC_WAIT(n) asm volatile("s_wait_asynccnt " #n ::: "memory")

// ---------------------------------------------------------------------------
// Kernel 1: mask-weighted spatial PSDs via V_WMMA_F32_16X16X4_F32 with
// double-buffered async Global->LDS staging.
// One wave owns 2 freq bins packed in a 16x16 fp32 tile (rows/cols 0-5 = f0,
// 8-13 = f1); K-dim = 4 time frames per WMMA; 4 (k,m) combos share B fragments.
// Partial PSDs per T-chunk go to workspace (deterministic two-pass reduction).
// ---------------------------------------------------------------------------
__global__ __launch_bounds__(256) void mvdr_psd_wmma(
    const float* __restrict__ masks,   // [K][M][T][F]
    const float* __restrict__ obr,     // [D][T][F]
    const float* __restrict__ obi,     // [D][T][F]
    float* __restrict__ part)          // [chunk][m][k][F][6][6][2]
{
  // [buf][f_local][t(padded)][c]: c = 2d -> xr(d), 2d+1 -> xi(d)   (d=0..7, 6..7 pad)
  __shared__ float sx[2][16 * TPAD * 16];
  // [buf][f_local][t(padded)][km]
  __shared__ float sm[2][16 * TPAD * 4];

  const int tid  = threadIdx.x;
  const int wave = tid >> 5;
  const int lane = tid & 31;
  const int row  = lane & 15;
  const int hi   = lane >> 4;

  const int fbase = blockIdx.x * 16;
  const int chunk = blockIdx.y;
  const int t0    = chunk * TCHUNK;
  const int tend  = (t0 + TCHUNK < T_) ? (t0 + TCHUNK) : T_;

  // A/B row<->(local f, mic) mapping: rows 0-7 -> f_local=2*wave, rows 8-15 -> +1
  const int afl    = 2 * wave + (row >> 3);
  const int ad     = row & 7;
  const bool avalid = (ad < 6);

  // Stage one tile into buffer `buf` with always-valid (clamped) addresses so
  // every wave issues exactly 16 async ops (uniform ASYNCcnt accounting).
  auto stage = [&](int buf, int tb) {
    const int tmax = tend - 1;
#pragma unroll
    for (int it = 0; it < 6; ++it) {              // Obs: 6*TT*16 = 1536 elems
      const int e  = tid + it * 256;
      const int fi = e & 15;
      const int ti = (e >> 4) & (TT - 1);
      const int d  = e >> 8;
      int f = fbase + fi; if (f > F_ - 1) f = F_ - 1;
      int t = tb + ti;    if (t > tmax)   t = tmax;
      const int gb = (d * TF_ + t * F_ + f) * 4;  // byte offset
      const int l  = (fi * TPAD + ti) * 16 + 2 * d;
      async_ld((unsigned)(size_t)&sx[buf][l],     obr, gb);
      async_ld((unsigned)(size_t)&sx[buf][l + 1], obi, gb);
    }
#pragma unroll
    for (int it = 0; it < 4; ++it) {              // masks: 4*TT*16 = 1024 elems
      const int e  = tid + it * 256;
      const int fi = e & 15;
      const int ti = (e >> 4) & (TT - 1);
      const int km = e >> 8;                      // km = k*2 + m
      int f = fbase + fi; if (f > F_ - 1) f = F_ - 1;
      int t = tb + ti;    if (t > tend - 1) t = tend - 1;
      const int gb = ((km * T_ + t) * F_ + f) * 4;
      async_ld((unsigned)(size_t)&sm[buf][(fi * TPAD + ti) * 4 + km], masks, gb);
    }
  };

  v8f cre[4], cim[4];
  const v8f vz = {0.f, 0.f, 0.f, 0.f, 0.f, 0.f, 0.f, 0.f};
#pragma unroll
  for (int i = 0; i < 4; ++i) { cre[i] = vz; cim[i] = vz; }

  stage(0, t0);                                   // prologue: 16 async in flight
  int cur = 0;
  for (int tb = t0; tb < tend; tb += TT) {
    const int  nxt = tb + TT;
    const bool hn  = (nxt < tend);
    if (hn) {
      stage(cur ^ 1, nxt);                        // 32 in flight
      ASYNC_WAIT(16);                             // current tile landed (in-order)
    } else {
      ASYNC_WAIT(0);
    }
    __syncthreads();                              // all waves' portions visible

    const float* xs = sx[cur];
    const float* ms = sm[cur];
#pragma unroll
    for (int tt = 0; tt < TT; tt += 4) {
      // this lane's two K-slots: t = tb + tt + 2*hi (+0, +1)
      const int tl = tt + 2 * hi;
      const int lx = (afl * TPAD + tl) * 16 + 2 * ad;
      float xr0 = xs[lx];
      float xi0 = xs[lx + 1];
      float xr1 = xs[lx + 16];
      float xi1 = xs[lx + 17];
      // zero padding rows AND clamped/out-of-range time slots (keeps A=0, B finite)
      const bool v0 = avalid && ((tb + tl)     < tend);
      const bool v1 = avalid && ((tb + tl + 1) < tend);
      if (!v0) { xr0 = 0.f; xi0 = 0.f; }
      if (!v1) { xr1 = 0.f; xi1 = 0.f; }

      const v2f brf  = { xr0,  xr1};              // B real
      const v2f byf  = { xi0,  xi1};              // B imag (not conjugated)
      const v2f bynf = {-xi0, -xi1};              // -B imag (conjugate term)

      const int lm = (afl * TPAD + tl) * 4;
#pragma unroll
      for (int km = 0; km < 4; ++km) {
        const float m0 = ms[lm + km];
        const float m1 = ms[lm + 4 + km];
        const v2f arf = {m0 * xr0, m1 * xr1};     // A real = mask*xr
        const v2f aif = {m0 * xi0, m1 * xi1};     // A imag = mask*xi
        // Re += Ar*Br + Ai*Bi ; Im += Ai*Br - Ar*Bi
        cre[km] = __builtin_amdgcn_wmma_f32_16x16x4_f32(
            false, arf, false, brf,  (short)0, cre[km], false, false);
        cre[km] = __builtin_amdgcn_wmma_f32_16x16x4_f32(
            false, aif, false, byf,  (short)0, cre[km], false, false);
        cim[km] = __builtin_amdgcn_wmma_f32_16x16x4_f32(
            false, aif, false, brf,  (short)0, cim[km], false, false);
        cim[km] = __builtin_amdgcn_wmma_f32_16x16x4_f32(
            false, arf, false, bynf, (short)0, cim[km], false, false);
      }
    }
    __syncthreads();                              // done reading `cur` before reuse
    cur ^= 1;
  }

  // ---- write partial PSDs ----
  // C layout: vector elem r, lanes 0-15 -> (M=r, N=lane); lanes 16-31 -> (M=r+8, N=lane-16)
  // f0 block: lanes 0-5 (hi=0, rows 0-5); f1 block: lanes 24-29 (hi=1, rows 8-13)
  const int cgrp = row >> 3;
  const int col  = row & 7;
  const int f    = fbase + 2 * wave + hi;
  if ((cgrp == hi) && (col < 6) && (f < F_)) {
    union { v8f v; float s[8]; } ure, uim;
#pragma unroll
    for (int km = 0; km < 4; ++km) {
      ure.v = cre[km];
      uim.v = cim[km];
      const int k = km >> 1;
      const int m = km & 1;
      const size_t base = ((((size_t)chunk * 2 + m) * 2 + k) * F_ + f) * 36;
#pragma unroll
      for (int r = 0; r < 6; ++r) {
        const size_t o = (base + r * 6 + col) * 2;
        part[o]     = ure.s[r];
        part[o + 1] = uim.s[r];
      }
    }
  }
}

// ---------------------------------------------------------------------------
// Kernel 2: reduce partials over T-chunks, solve interf \ target (6x6 complex,
// no-pivot Gauss-Jordan: interference PSD is Hermitian PD), trace-normalize,
// emit conj(beamformer)/lam for the reference channel. One thread per (k,f).
// ---------------------------------------------------------------------------
__global__ __launch_bounds__(64) void mvdr_reduce_solve(
    const float* __restrict__ part,
    const int*   __restrict__ refp,
    float* __restrict__ wout)          // [K][F][6][2] = conj(w)/lam
{
  const int idx = blockIdx.x * blockDim.x + threadIdx.x;
  if (idx >= KSPK * F_) return;
  const int k = idx / F_;
  const int f = idx - k * F_;
  const int ref = refp[0];

  float Tre[6][6], Tim[6][6], Bre[6][6], Bim[6][6];
#pragma unroll
  for (int d = 0; d < 6; ++d)
#pragma unroll
    for (int c = 0; c < 6; ++c) { Tre[d][c] = Tim[d][c] = Bre[d][c] = Bim[d][c] = 0.f; }

  for (int ch = 0; ch < NCHUNK; ++ch) {
    const float* p0 = part + ((((size_t)ch * 2 + 0) * 2 + k) * F_ + f) * 72; // target (m=0)
    const float* p1 = part + ((((size_t)ch * 2 + 1) * 2 + k) * F_ + f) * 72; // interference
#pragma unroll
    for (int d = 0; d < 6; ++d)
#pragma unroll
      for (int c = 0; c < 6; ++c) {
        const int e = d * 6 + c;
        Tre[d][c] += p0[2 * e];
        Tim[d][c] += p0[2 * e + 1];
        Bre[d][c] += p1[2 * e];
        Bim[d][c] += p1[2 * e + 1];
      }
  }

  // Gauss-Jordan: reduce B to I, T becomes phi = B^{-1} T
#pragma unroll
  for (int p = 0; p < 6; ++p) {
    const float pr = Bre[p][p], pi = Bim[p][p];
    const float inv = 1.0f / (pr * pr + pi * pi);
    const float ir = pr * inv, ii = -pi * inv;
#pragma unroll
    for (int c = 0; c < 6; ++c) {
      float xr = Bre[p][c], xi = Bim[p][c];
      Bre[p][c] = xr * ir - xi * ii;
      Bim[p][c] = xr * ii + xi * ir;
      xr = Tre[p][c]; xi = Tim[p][c];
      Tre[p][c] = xr * ir - xi * ii;
      Tim[p][c] = xr * ii + xi * ir;
    }
#pragma unroll
    for (int r = 0; r < 6; ++r) {
      if (r == p) continue;
      const float fr = Bre[r][p], fi = Bim[r][p];
#pragma unroll
      for (int c = 0; c < 6; ++c) {
        const float bpr = Bre[p][c], bpi = Bim[p][c];
        Bre[r][c] -= fr * bpr - fi * bpi;
        Bim[r][c] -= fr * bpi + fi * bpr;
        const float tpr = Tre[p][c], tpi = Tim[p][c];
        Tre[r][c] -= fr * tpr - fi * tpi;
        Tim[r][c] -= fr * tpi + fi * tpr;
      }
    }
  }

  float lr = 0.f;
#pragma unroll
  for (int d = 0; d < 6; ++d) lr += Tre[d][d];
  const float lam = fmaxf(lr, 1.17549435e-38f);   // jnp.finfo(f32).tiny
  const float s = 1.0f / lam;

  const size_t wb = ((size_t)k * F_ + f) * 6;
#pragma unroll
  for (int d = 0; d < 6; ++d) {
    float wr = 0.f, wi = 0.f;
#pragma unroll
    for (int c = 0; c < 6; ++c)
      if (c == ref) { wr = Tre[d][c]; wi = Tim[d][c]; }
    wout[(wb + d) * 2]     =  wr * s;   // conj applied: store (re, -im)
    wout[(wb + d) * 2 + 1] = -wi * s;
  }
}

// ---------------------------------------------------------------------------
// Kernel 3: enh[k,t,f] = sum_d wc[k,f,d] * x[d,t,f]   (wc already conjugated)
// Streaming-bound; coalesced over f. Output interleaved complex64.
// ---------------------------------------------------------------------------
__global__ __launch_bounds__(256) void mvdr_apply(
    const float* __restrict__ obr,
    const float* __restrict__ obi,
    const float* __restrict__ wc,
    float* __restrict__ out)
{
  const int idx = blockIdx.x * 256 + threadIdx.x;
  if (idx >= KSPK * TF_) return;
  const int k   = idx / TF_;
  const int rem = idx - k * TF_;        // = t*F + f
  const int f   = rem % F_;

  const float* w = wc + ((size_t)k * F_ + f) * 12;
  float er = 0.f, ei = 0.f;
#pragma unroll
  for (int d = 0; d < 6; ++d) {
    const float xr = obr[d * TF_ + rem];
    const float xi = obi[d * TF_ + rem];
    const float wr = w[2 * d];
    const float wi = w[2 * d + 1];
    er += wr * xr - wi * xi;
    ei += wr * xi + wi * xr;
  }
  out[(size_t)idx * 2]     = er;
  out[(size_t)idx * 2 + 1] = ei;
}

extern "C" void kernel_launch(void* const* d_in, const int* in_sizes, int n_in,
                              void* d_out, int out_size, void* d_ws, size_t ws_size,
                              hipStream_t stream) {
  const float* masks = (const float*)d_in[0];   // [2][2][3000][513]
  const float* obr   = (const float*)d_in[1];   // [6][3000][513]
  const float* obi   = (const float*)d_in[2];   // [6][3000][513]
  const int*   refp  = (const int*)d_in[3];     // scalar
  float* out  = (float*)d_out;                  // [2][3000][513] complex64 (interleaved)
  float* part = (float*)d_ws;                   // ~4.73 MB partial PSDs
  float* wcon = part + PARTIAL_FLOATS;          // ~49 KB conj beamformer

  dim3 g1((F_ + 15) / 16, NCHUNK);              // 33 x 8 workgroups
  mvdr_psd_wmma<<<g1, 256, 0, stream>>>(masks, obr, obi, part);

  const int nkf = KSPK * F_;                    // 1026 solves
  mvdr_reduce_solve<<<(nkf + 63) / 64, 64, 0, stream>>>(part, refp, wcon);

  const int ntot = KSPK * TF_;                  // 3,078,000 outputs
  mvdr_apply<<<(ntot + 255) / 256, 256, 0, stream>>>(obr, obi, wcon, out);
}